// HeatmapDecoder_58858231824720
// MI455X (gfx1250) — compile-verified
//
#include <hip/hip_runtime.h>
#include <hip/hip_bf16.h>

// ---------------- problem constants ----------------
#define BATCH   32
#define HH      256
#define WW      256
#define HW      65536          // 256*256
#define NJ      8
#define KTOP    100
#define KPAD    128            // padded stride for per-plane topk arrays
#define NPLANES 288            // 32 hm planes + 256 hm_hp planes
#define PADM    112            // 100 padded to 7*16 for WMMA tiling
#define THRESH  0.1f
#define NEGV    (-10000.0f)

// ---------------- output layout (floats, concatenated in return order) ----
#define OFF_BBOX   0                             // (32,100,4)   = 12800
#define OFF_SCORES (OFF_BBOX   + BATCH*KTOP*4)   // (32,100,1) = 3200
#define OFF_KPS    (OFF_SCORES + BATCH*KTOP)     // (32,100,16)= 51200
#define OFF_CLSES  (OFF_KPS    + BATCH*KTOP*16)  // (32,100,1) = 3200
#define OFF_SCALE  (OFF_CLSES  + BATCH*KTOP)     // (32,100,3) = 9600
#define OFF_DISP   (OFF_SCALE  + BATCH*KTOP*3)   // (32,100,16)= 51200
#define OFF_KHM    (OFF_DISP   + BATCH*KTOP*16)  // (32,100,16)= 51200

typedef __attribute__((ext_vector_type(2))) float v2f;
typedef __attribute__((ext_vector_type(8))) float v8f;

// order-preserving float->uint key (larger float -> larger key)
__device__ __forceinline__ unsigned order_key(float x) {
    unsigned u = __float_as_uint(x);
    return (u & 0x80000000u) ? ~u : (u | 0x80000000u);
}
__device__ __forceinline__ float unorder_key(unsigned k) {
    unsigned u = (k & 0x80000000u) ? (k ^ 0x80000000u) : ~k;
    return __uint_as_float(u);
}

// c >= all 3 neighbors in row r (branch-free: clamped address + masked compare)
__device__ __forceinline__ bool nbr3_ge(const float* __restrict__ r, int x, float c) {
    bool ok = c >= r[x];
    float vl = r[x > 0      ? x - 1 : x];
    float vr = r[x < WW - 1 ? x + 1 : x];
    ok = ok && (x == 0      || c >= vl);
    ok = ok && (x == WW - 1 || c >= vr);
    return ok;
}
// c >= left/right neighbors only (center row)
__device__ __forceinline__ bool nbr2_ge(const float* __restrict__ r, int x, float c) {
    float vl = r[x > 0      ? x - 1 : x];
    float vr = r[x < WW - 1 ? x + 1 : x];
    bool ok  =  (x == 0      || c >= vl);
    ok = ok && (x == WW - 1 || c >= vr);
    return ok;
}

// ============================================================================
// Kernel 1: per-plane sigmoid + 3x3 NMS + exact sorted top-100.
// Streams the plane with an LDS 4-row ring (1 global load/elem/pass, one
// barrier per row).  Two passes: histogram-select threshold, then collect.
// The 256KB plane lives in the 192MB L2, so the second pass is L2-resident.
// ============================================================================
__global__ void __launch_bounds__(256)
topk_plane_kernel(const float* __restrict__ hm, const float* __restrict__ hm_hp,
                  float* __restrict__ sc_out, int* __restrict__ ind_out) {
    const int plane = blockIdx.x;
    const float* __restrict__ src =
        (plane < BATCH) ? (hm + (size_t)plane * HW)
                        : (hm_hp + (size_t)(plane - BATCH) * HW);
    __shared__ unsigned hist[2048];
    __shared__ unsigned long long cand[1024];
    __shared__ float rows[4][WW];
    __shared__ unsigned cnt;
    __shared__ unsigned tbin;

    const int t = threadIdx.x;             // thread == column x
    for (int i = t; i < 2048; i += 256) hist[i] = 0u;
    if (t == 0) { cnt = 0u; tbin = 0u; }
    rows[0][t] = src[t];
    __syncthreads();

    // ---- pass A: histogram of NMS survivors ----
    for (int y = 0; y < HH; ++y) {
        if (y + 1 < HH) rows[(y + 1) & 3][t] = src[((y + 1) << 8) + t];
        __syncthreads();
        const float* rm = rows[(y + 3) & 3];
        const float* rc = rows[y & 3];
        const float* rp = rows[(y + 1) & 3];
        float c = rc[t];
        bool ok = nbr2_ge(rc, t, c);
        if (y > 0)      ok = ok && nbr3_ge(rm, t, c);
        if (y < HH - 1) ok = ok && nbr3_ge(rp, t, c);
        if (ok) atomicAdd(&hist[order_key(c) >> 21], 1u);
    }
    __syncthreads();

    if (t == 0) {
        unsigned cum = 0; int bin = 0;
        for (int b = 2047; b >= 0; --b) {
            cum += hist[b];
            if (cum >= (unsigned)KTOP) { bin = b; break; }
        }
        tbin = (unsigned)bin;
    }
    rows[0][t] = src[t];
    __syncthreads();
    const unsigned T = tbin;

    // ---- pass B: collect candidates (key, ~idx): u64-descending sort gives
    // score-desc / index-asc order (matches jax top_k tie-break) ----
    for (int y = 0; y < HH; ++y) {
        if (y + 1 < HH) rows[(y + 1) & 3][t] = src[((y + 1) << 8) + t];
        __syncthreads();
        const float* rm = rows[(y + 3) & 3];
        const float* rc = rows[y & 3];
        const float* rp = rows[(y + 1) & 3];
        float c = rc[t];
        bool ok = nbr2_ge(rc, t, c);
        if (y > 0)      ok = ok && nbr3_ge(rm, t, c);
        if (y < HH - 1) ok = ok && nbr3_ge(rp, t, c);
        if (ok) {
            unsigned key = order_key(c);
            if ((key >> 21) >= T) {
                unsigned pos = atomicAdd(&cnt, 1u);
                if (pos < 1024u) {
                    unsigned i = (unsigned)((y << 8) + t);
                    cand[pos] = ((unsigned long long)key << 32) | (unsigned)(~i);
                }
            }
        }
    }
    __syncthreads();
    unsigned n = cnt; if (n > 1024u) n = 1024u;
    for (int i = t; i < 1024; i += 256)
        if ((unsigned)i >= n) cand[i] = 0ull;
    __syncthreads();

    // ---- bitonic sort, descending, 1024 elems, 256 threads ----
    for (int k = 2; k <= 1024; k <<= 1) {
        for (int j = k >> 1; j > 0; j >>= 1) {
            for (int i = t; i < 1024; i += 256) {
                int ixj = i ^ j;
                if (ixj > i) {
                    unsigned long long a = cand[i], b = cand[ixj];
                    bool up = ((i & k) == 0);
                    bool sw = up ? (a < b) : (a > b);
                    if (sw) { cand[i] = b; cand[ixj] = a; }
                }
            }
            __syncthreads();
        }
    }

    if (t < KTOP) {
        unsigned long long e = cand[t];
        unsigned key = (unsigned)(e >> 32);
        float score = 0.0f;
        unsigned idx = 0u;
        if (key != 0u) {
            idx = ~(unsigned)(e & 0xFFFFFFFFull);
            float x = unorder_key(key);
            score = 1.0f / (1.0f + __expf(-x));
        }
        sc_out[plane * KPAD + t]  = score;
        ind_out[plane * KPAD + t] = (int)idx;
    }
}

// ============================================================================
// Kernel 2: K=100 gathers per batch -> bboxes/scores/clses/scale/disp outputs
// plus workspace arrays feeding the WMMA assignment kernel.
// ============================================================================
__global__ void __launch_bounds__(256)
gather_kernel(const float* __restrict__ wh, const float* __restrict__ kps,
              const float* __restrict__ reg, const float* __restrict__ hp_off,
              const float* __restrict__ scale,
              const float* __restrict__ sc, const int* __restrict__ ind,
              float* __restrict__ out,
              float* __restrict__ kpx, float* __restrict__ kpy,
              float* __restrict__ hx, float* __restrict__ hy,
              float* __restrict__ hs) {
    const int b = blockIdx.x;
    const int t = threadIdx.x;

    if (t < KTOP) {
        const int k = t;
        const int iv = ind[b * KPAD + k];
        const float s = sc[b * KPAD + k];
        const float ysf = (float)(iv >> 8), xsf = (float)(iv & 255);
        const float rx = reg[(size_t)(b * 2 + 0) * HW + iv];
        const float ry = reg[(size_t)(b * 2 + 1) * HW + iv];
        const float w0 = wh [(size_t)(b * 2 + 0) * HW + iv];
        const float w1 = wh [(size_t)(b * 2 + 1) * HW + iv];
        const float xr = xsf + rx, yr = ysf + ry;
        const int bo = (b * KTOP + k) * 4;
        out[OFF_BBOX + bo + 0] = xr - w0 * 0.5f;
        out[OFF_BBOX + bo + 1] = yr - w1 * 0.5f;
        out[OFF_BBOX + bo + 2] = xr + w0 * 0.5f;
        out[OFF_BBOX + bo + 3] = yr + w1 * 0.5f;
        out[OFF_SCORES + b * KTOP + k] = s;
        out[OFF_CLSES  + b * KTOP + k] = 0.0f;   // single class
        const int so = (b * KTOP + k) * 3;
        out[OFF_SCALE + so + 0] = scale[(size_t)(b * 3 + 0) * HW + iv];
        out[OFF_SCALE + so + 1] = scale[(size_t)(b * 3 + 1) * HW + iv];
        out[OFF_SCALE + so + 2] = scale[(size_t)(b * 3 + 2) * HW + iv];
    }

    // keypoint displacements: (k, j) pairs
    for (int p = t; p < KTOP * NJ; p += 256) {
        const int k = p >> 3, j = p & 7;
        const int iv = ind[b * KPAD + k];
        const float ysf = (float)(iv >> 8), xsf = (float)(iv & 255);
        const float vx = kps[(size_t)(b * 16 + j * 2 + 0) * HW + iv] + xsf;
        const float vy = kps[(size_t)(b * 16 + j * 2 + 1) * HW + iv] + ysf;
        const int off = (b * KTOP + k) * 16 + j * 2;
        out[OFF_DISP + off]     = vx;
        out[OFF_DISP + off + 1] = vy;
        kpx[(b * NJ + j) * KPAD + k] = vx;
        kpy[(b * NJ + j) * KPAD + k] = vy;
    }

    // heatmap joint candidates: (j, k)
    for (int p = t; p < NJ * KTOP; p += 256) {
        const int j = p / KTOP, k = p - j * KTOP;
        const int pl = BATCH + b * NJ + j;
        const int iv = ind[pl * KPAD + k];
        const float s = sc[pl * KPAD + k];
        const float ysf = (float)(iv >> 8), xsf = (float)(iv & 255);
        const float ox = hp_off[(size_t)(b * 2 + 0) * HW + iv];
        const float oy = hp_off[(size_t)(b * 2 + 1) * HW + iv];
        const bool m = s > THRESH;
        const int o = (b * NJ + j) * KPAD + k;
        hx[o] = m ? (xsf + ox) : NEGV;
        hy[o] = m ? (ysf + oy) : NEGV;
        hs[o] = m ? s : -1.0f;
    }
}

// ============================================================================
// Kernel 3: 100x100 squared-distance matrix per (b,joint) via
// V_WMMA_F32_16X16X4_F32:  dist^2 = A x B with
//   A row m = (kx, ky, kx^2+ky^2, 1),  B col n = (-2hx, -2hy, 1, hx^2+hy^2)
// Operands are pre-arranged in LDS in WMMA lane order (low 16 lanes = K0/K1,
// high 16 lanes = K2/K3) so each lane does ONE unconditional ds_load_b64 per
// operand -> no exec-mask branches between WMMAs.  Column min/argmin is a
// 16-lane shfl_xor reduce matching the C/D layout (lane = column, vgpr = row).
// One wave per block -> EXEC all ones at the WMMA (ISA requirement).
// ============================================================================
__global__ void __launch_bounds__(32)
assign_kernel(const float* __restrict__ kpx, const float* __restrict__ kpy,
              const float* __restrict__ hxw, const float* __restrict__ hyw,
              const float* __restrict__ hsw,
              float* __restrict__ out) {
    const int bj = blockIdx.x;
    const int b = bj >> 3, j = bj & 7;
    const int lane = threadIdx.x;

    __shared__ v2f   aop[2 * PADM];    // [0..111]=(kx,ky)      [112..223]=(kn,1)
    __shared__ v2f   bop[2 * PADM];    // [0..111]=(-2hx,-2hy)  [112..223]=(1,hn)
    __shared__ float akx[PADM], aky[PADM];
    __shared__ float bx[PADM], by[PADM], bs[PADM];
    __shared__ float mind[PADM];
    __shared__ int   mini[PADM];

    for (int i = lane; i < PADM; i += 32) {
        float kx = 0.f, ky = 0.f;
        if (i < KTOP) { kx = kpx[bj * KPAD + i]; ky = kpy[bj * KPAD + i]; }
        akx[i] = kx; aky[i] = ky;
        aop[i]        = (v2f){kx, ky};
        aop[PADM + i] = (v2f){kx * kx + ky * ky, 1.0f};
        float hx = 0.f, hy = 0.f, nn = 3.0e30f, s = -1.f;   // padded cols never win
        if (i < KTOP) {
            hx = hxw[bj * KPAD + i]; hy = hyw[bj * KPAD + i];
            s  = hsw[bj * KPAD + i]; nn = hx * hx + hy * hy;
        }
        bx[i] = hx; by[i] = hy; bs[i] = s;
        bop[i]        = (v2f){-2.0f * hx, -2.0f * hy};
        bop[PADM + i] = (v2f){1.0f, nn};
    }
    __syncthreads();

    const bool hi   = lane >= 16;
    const int  l16  = lane & 15;
    const int  half = hi ? PADM : 0;

    for (int mt = 0; mt < 7; ++mt) {
        const v2f A = aop[half + mt * 16 + l16];
        float best[8]; int bcol[8];
#pragma unroll
        for (int v = 0; v < 8; ++v) { best[v] = 3.4e38f; bcol[v] = 1 << 30; }

#pragma unroll
        for (int nt = 0; nt < 7; ++nt) {
            const int col = nt * 16 + l16;
            const v2f Bv = bop[half + col];
            v8f C = {};
            v8f D = __builtin_amdgcn_wmma_f32_16x16x4_f32(
                        false, A, false, Bv, (short)0, C, false, false);
#pragma unroll
            for (int v = 0; v < 8; ++v) {
                float d2 = D[v];
                bool better = (d2 < best[v]) | ((d2 == best[v]) & (col < bcol[v]));
                best[v] = better ? d2  : best[v];
                bcol[v] = better ? col : bcol[v];
            }
        }
        // min/argmin across the 16 lanes of each half (lane = column)
#pragma unroll
        for (int v = 0; v < 8; ++v) {
            float d2 = best[v]; int c = bcol[v];
#pragma unroll
            for (int off = 1; off < 16; off <<= 1) {
                float od = __shfl_xor(d2, off, 32);
                int   oc = __shfl_xor(c,  off, 32);
                bool better = (od < d2) | ((od == d2) & (oc < c));
                d2 = better ? od : d2;
                c  = better ? oc : c;
            }
            const int row = mt * 16 + (hi ? 8 : 0) + v;
            if (l16 == 0 && row < KTOP) {
                mind[row] = sqrtf(fmaxf(d2, 0.0f));
                mini[row] = c;
            }
        }
    }
    __syncthreads();

    for (int k = lane; k < KTOP; k += 32) {
        const int bo = (b * KTOP + k) * 4;
        const float left   = out[OFF_BBOX + bo + 0];
        const float top    = out[OFF_BBOX + bo + 1];
        const float right  = out[OFF_BBOX + bo + 2];
        const float bottom = out[OFF_BBOX + bo + 3];
        const float score  = out[OFF_SCORES + b * KTOP + k];
        const int   mi = mini[k];
        const float md = mind[k];
        const float hx = bx[mi], hy = by[mi], hsc = bs[mi];
        const float kx = akx[k], ky = aky[k];
        const float diag = fmaxf(bottom - top, right - left);

        const bool m1 = (hx < left) || (hx > right) || (hy < top) || (hy > bottom)
                     || (hsc < THRESH) || (md > diag * 0.3f);
        const int ko = (b * KTOP + k) * 16 + j * 2;
        out[OFF_KPS + ko]     = m1 ? kx : hx;
        out[OFF_KPS + ko + 1] = m1 ? ky : hy;

        const bool m2 = (hx > 0.8f * left) && (hx < 1.2f * right)
                     && (hy > 0.8f * top) && (hy < 1.2f * bottom)
                     && (hsc > THRESH) && (md < diag * 0.5f) && (score > THRESH);
        const float fx = m2 ? hx : NEGV;
        const float fy = m2 ? hy : NEGV;
        const bool valid = (fx != NEGV) && (fy != NEGV);
        out[OFF_KHM + ko]     = valid ? fx : NEGV;
        out[OFF_KHM + ko + 1] = valid ? fy : NEGV;
    }
}

// ============================================================================
extern "C" void kernel_launch(void* const* d_in, const int* in_sizes, int n_in,
                              void* d_out, int out_size, void* d_ws, size_t ws_size,
                              hipStream_t stream) {
    const float* hm     = (const float*)d_in[0];
    const float* wh     = (const float*)d_in[1];
    const float* kps    = (const float*)d_in[2];
    const float* reg    = (const float*)d_in[3];
    const float* hm_hp  = (const float*)d_in[4];
    const float* hp_off = (const float*)d_in[5];
    const float* scale  = (const float*)d_in[6];
    float* out = (float*)d_out;

    // workspace carve-up (floats): ~950 KB total
    float* sc  = (float*)d_ws;                       // NPLANES*KPAD
    int*   ind = (int*)(sc + NPLANES * KPAD);        // NPLANES*KPAD
    float* kpx = (float*)(ind + NPLANES * KPAD);     // 256*KPAD
    float* kpy = kpx + 256 * KPAD;
    float* hx  = kpy + 256 * KPAD;
    float* hy  = hx  + 256 * KPAD;
    float* hs  = hy  + 256 * KPAD;

    topk_plane_kernel<<<NPLANES, 256, 0, stream>>>(hm, hm_hp, sc, ind);
    gather_kernel<<<BATCH, 256, 0, stream>>>(wh, kps, reg, hp_off, scale,
                                             sc, ind, out, kpx, kpy, hx, hy, hs);
    assign_kernel<<<BATCH * NJ, 32, 0, stream>>>(kpx, kpy, hx, hy, hs, out);
}